// NeRFRenderer_3831110828163
// MI455X (gfx1250) — compile-verified
//
#include <hip/hip_runtime.h>

// ---------------------------------------------------------------------------
// NeRF ray-march renderer for MI455X (gfx1250, wave32).
//
// Roofline: ~1 GFLOP total, 32 MB grid (L2-resident in 192 MB L2),
// ~1.07 GB of divergent L2 gather traffic -> latency/L2-bound, not compute.
// Strategy: one ray per thread (2048 wave32 waves in flight), 8x B128 corner
// gathers per step, software-pipelined cell computation with
// global_prefetch_b8 of the NEXT step's 4 cachelines, early ray termination.
// WMMA intentionally unused: gathered [1x8]x[8x4] trilerp has no profitable
// mapping onto 16x16 WMMA lane layouts; shuffle cost > FLOPs saved.
// ---------------------------------------------------------------------------

namespace {
constexpr int   kSteps      = 128;
constexpr int   kGrid       = 128;
constexpr float kBound      = 1.0f;
constexpr float kMinNear    = 0.05f;
constexpr float kTThresh    = 1e-4f;
constexpr float kSigmaScale = 25.0f;
constexpr float kBgColor    = 1.0f;
constexpr float kPosScale   = (kGrid - 1) * 0.5f;          // (G-1)/(2*BOUND) = 63.5
constexpr float kPosMax     = (float)(kGrid - 1) - 1e-5f;  // clip upper bound
}  // namespace

__device__ __forceinline__ float safe_inv(float d) {
  // matches reference: 1.0 / where(|d| < 1e-9, 1e-9, d)
  return 1.0f / (fabsf(d) < 1e-9f ? 1e-9f : d);
}

__device__ __forceinline__ float lerpf(float a, float b, float f) {
  // a*(1-f) + b*f
  return fmaf(f, b, fmaf(-f, a, a));
}

struct CellCoords {
  int   b00;  // float4-index of (x0, y0, 0)
  int   z0;
  float fx, fy, fz;
};

__device__ __forceinline__ CellCoords cell_at(float t,
                                              float osx, float dsx,
                                              float osy, float dsy,
                                              float osz, float dsz) {
  CellCoords c;
  const float px = fminf(fmaxf(fmaf(t, dsx, osx), 0.0f), kPosMax);
  const float py = fminf(fmaxf(fmaf(t, dsy, osy), 0.0f), kPosMax);
  const float pz = fminf(fmaxf(fmaf(t, dsz, osz), 0.0f), kPosMax);
  const int x0 = (int)px, y0 = (int)py, z0 = (int)pz;  // pos >= 0 -> floor == trunc
  c.b00 = (x0 * kGrid + y0) * kGrid;
  c.z0  = z0;
  c.fx  = px - (float)x0;
  c.fy  = py - (float)y0;
  c.fz  = pz - (float)z0;
  return c;
}

__global__ __launch_bounds__(256) void nerf_march_kernel(
    const float* __restrict__ rays_o,   // [N,3]
    const float* __restrict__ rays_d,   // [N,3]
    const float4* __restrict__ grid,    // [128,128,128] of float4 (4 feats)
    float* __restrict__ out_img,        // [N,3]
    float* __restrict__ out_depth,      // [N]
    float* __restrict__ out_wsum,       // [N]
    int n_rays) {
  const int r = blockIdx.x * blockDim.x + threadIdx.x;
  if (r >= n_rays) return;

  const float ox = rays_o[3 * r + 0];
  const float oy = rays_o[3 * r + 1];
  const float oz = rays_o[3 * r + 2];
  const float dx = rays_d[3 * r + 0];
  const float dy = rays_d[3 * r + 1];
  const float dz = rays_d[3 * r + 2];

  // ---- near / far (slab test, reference semantics) ----
  const float ivx = safe_inv(dx), ivy = safe_inv(dy), ivz = safe_inv(dz);
  const float t1x = (-kBound - ox) * ivx, t2x = (kBound - ox) * ivx;
  const float t1y = (-kBound - oy) * ivy, t2y = (kBound - oy) * ivy;
  const float t1z = (-kBound - oz) * ivz, t2z = (kBound - oz) * ivz;
  const float tmin = fmaxf(fmaxf(fminf(t1x, t2x), fminf(t1y, t2y)), fminf(t1z, t2z));
  const float tmax = fminf(fminf(fmaxf(t1x, t2x), fmaxf(t1y, t2y)), fmaxf(t1z, t2z));
  const float near = fmaxf(tmin, kMinNear);
  const float far  = fmaxf(tmax, near + 1e-5f);
  const float dt   = (far - near) * (1.0f / kSteps);

  // pos = (o + t*d + BOUND) * kPosScale, folded into one fma per axis
  const float osx = (ox + kBound) * kPosScale, dsx = dx * kPosScale;
  const float osy = (oy + kBound) * kPosScale, dsy = dy * kPosScale;
  const float osz = (oz + kBound) * kPosScale, dsz = dz * kPosScale;

  float T     = 1.0f;  // transmittance BEFORE current sample
  float wsum  = 0.0f;
  float depth = 0.0f;
  float accR = 0.0f, accG = 0.0f, accB = 0.0f;

  float t = fmaf(0.5f, dt, near);

  // Software pipeline: cell coords for step s are computed during step s-1
  // (right after its gathers were issued), and the 4 cachelines are
  // prefetched via global_prefetch_b8 so step s's gathers hit near caches.
  CellCoords cur = cell_at(t, osx, dsx, osy, dsy, osz, dsz);

  for (int s = 0; s < kSteps; ++s) {
    // ---- issue the 8 B128 corner gathers for the CURRENT cell first ----
    const int b00 = cur.b00;                 // (x0,y0,*)
    const int b01 = b00 + kGrid;             // (x0,y1,*)
    const int b10 = b00 + kGrid * kGrid;     // (x1,y0,*)
    const int b11 = b10 + kGrid;             // (x1,y1,*)
    const int z0  = cur.z0;
    const float4 c000 = grid[b00 + z0], c001 = grid[b00 + z0 + 1];
    const float4 c010 = grid[b01 + z0], c011 = grid[b01 + z0 + 1];
    const float4 c100 = grid[b10 + z0], c101 = grid[b10 + z0 + 1];
    const float4 c110 = grid[b11 + z0], c111 = grid[b11 + z0 + 1];

    // ---- while gathers are in flight: compute & prefetch NEXT cell ----
    const float t_next = t + dt;
    const CellCoords nxt = cell_at(t_next, osx, dsx, osy, dsy, osz, dsz);
    {
      const float4* pf = grid + (nxt.b00 + nxt.z0);
      __builtin_prefetch(pf,                         0, 3);  // (x0,y0) z-pair
      __builtin_prefetch(pf + kGrid,                 0, 3);  // (x0,y1) z-pair
      __builtin_prefetch(pf + kGrid * kGrid,         0, 3);  // (x1,y0) z-pair
      __builtin_prefetch(pf + kGrid * kGrid + kGrid, 0, 3);  // (x1,y1) z-pair
    }

    // ---- trilerp: z, then y, then x (reference order), per channel ----
    const float fx = cur.fx, fy = cur.fy, fz = cur.fz;
    float f0, f1, f2, f3;
    {
      const float a00 = lerpf(c000.x, c001.x, fz), a01 = lerpf(c010.x, c011.x, fz);
      const float a10 = lerpf(c100.x, c101.x, fz), a11 = lerpf(c110.x, c111.x, fz);
      f0 = lerpf(lerpf(a00, a01, fy), lerpf(a10, a11, fy), fx);
    }
    {
      const float a00 = lerpf(c000.y, c001.y, fz), a01 = lerpf(c010.y, c011.y, fz);
      const float a10 = lerpf(c100.y, c101.y, fz), a11 = lerpf(c110.y, c111.y, fz);
      f1 = lerpf(lerpf(a00, a01, fy), lerpf(a10, a11, fy), fx);
    }
    {
      const float a00 = lerpf(c000.z, c001.z, fz), a01 = lerpf(c010.z, c011.z, fz);
      const float a10 = lerpf(c100.z, c101.z, fz), a11 = lerpf(c110.z, c111.z, fz);
      f2 = lerpf(lerpf(a00, a01, fy), lerpf(a10, a11, fy), fx);
    }
    {
      const float a00 = lerpf(c000.w, c001.w, fz), a01 = lerpf(c010.w, c011.w, fz);
      const float a10 = lerpf(c100.w, c101.w, fz), a11 = lerpf(c110.w, c111.w, fz);
      f3 = lerpf(lerpf(a00, a01, fy), lerpf(a10, a11, fy), fx);
    }

    // ---- volume rendering accumulation ----
    const float sigma = f0 * kSigmaScale;
    const float sdt   = sigma * dt;
    const float att   = __expf(-sdt);  // bare v_exp_f32 (TRANS, co-executes)
    const float alpha = 1.0f - att;
    const float w     = (T > kTThresh) ? (alpha * T) : 0.0f;

    wsum  += w;
    depth  = fmaf(w, t, depth);
    accR   = fmaf(w, f1, accR);
    accG   = fmaf(w, f2, accG);
    accB   = fmaf(w, f3, accB);

    T *= att;  // transmittance for the NEXT sample
    // sigma >= 0 => T monotone non-increasing: once below threshold every
    // remaining weight is exactly 0 -> terminate the ray (deterministic).
    if (T <= kTThresh) break;

    // rotate pipeline state
    cur = nxt;
    t   = t_next;
  }

  const float bg = (1.0f - wsum) * kBgColor;
  out_img[3 * r + 0] = accR + bg;
  out_img[3 * r + 1] = accG + bg;
  out_img[3 * r + 2] = accB + bg;
  out_depth[r] = depth;
  out_wsum[r]  = wsum;
}

extern "C" void kernel_launch(void* const* d_in, const int* in_sizes, int n_in,
                              void* d_out, int out_size, void* d_ws, size_t ws_size,
                              hipStream_t stream) {
  (void)n_in; (void)d_ws; (void)ws_size; (void)out_size;

  const float*  rays_o = (const float*)d_in[0];   // [N,3]
  const float*  rays_d = (const float*)d_in[1];   // [N,3]
  const float4* grid   = (const float4*)d_in[2];  // [128,128,128,4] floats

  const int n_rays = in_sizes[0] / 3;  // 65536

  // d_out layout: image [N,3] ‖ depth [N] ‖ weights_sum [N]
  float* out_img   = (float*)d_out;
  float* out_depth = out_img + (size_t)3 * n_rays;
  float* out_wsum  = out_depth + n_rays;

  const int block = 256;  // 8 wave32 waves per block
  const int grid_dim = (n_rays + block - 1) / block;
  nerf_march_kernel<<<grid_dim, block, 0, stream>>>(
      rays_o, rays_d, grid, out_img, out_depth, out_wsum, n_rays);
}